// DistanceScore_1589137899812
// MI455X (gfx1250) — compile-verified
//
#include <hip/hip_runtime.h>
#include <math.h>

#define N1 8192
#define N2 8192
#define DIMK 64
#define TOPK 32
#define ROWS_PER_BLK 16
#define CHUNK 512
#define DPITCH (CHUNK + 4)
#define WAVES 8
#define TPB 256
#define TILES_PER_WAVE (CHUNK / 16 / WAVES)   // 4

typedef __attribute__((ext_vector_type(2))) float v2f;
typedef __attribute__((ext_vector_type(8))) float v8f;

// Bitonic sort of 32 (value,index) pairs across a wave32, ascending by value.
__device__ __forceinline__ void bitonic_sort_asc(float &v, int &i) {
    const int lane = threadIdx.x & 31;
#pragma unroll
    for (int k = 2; k <= 32; k <<= 1) {
#pragma unroll
        for (int j = k >> 1; j >= 1; j >>= 1) {
            float ov = __shfl_xor(v, j, 32);
            int   oi = __shfl_xor(i, j, 32);
            const bool asc   = ((lane & k) == 0);
            const bool lower = ((lane & j) == 0);
            const bool take  = asc ? (lower ? (ov < v) : (ov > v))
                                   : (lower ? (ov > v) : (ov < v));
            if (take) { v = ov; i = oi; }
        }
    }
}

// Resident list rv sorted descending (lane i = i-th largest). Batch bv sorted
// ascending. Elementwise max gives a bitonic sequence holding the top-32 of
// the union; a 5-stage descending bitonic merge re-sorts it.
__device__ __forceinline__ void topk_merge_desc(float &rv, int &ri, float bv, int bi) {
    if (bv > rv) { rv = bv; ri = bi; }
    const int lane = threadIdx.x & 31;
#pragma unroll
    for (int j = 16; j >= 1; j >>= 1) {
        float ov = __shfl_xor(rv, j, 32);
        int   oi = __shfl_xor(ri, j, 32);
        const bool lower = ((lane & j) == 0);
        const bool take  = lower ? (ov > rv) : (ov < rv);
        if (take) { rv = ov; ri = oi; }
    }
}

__global__ __launch_bounds__(TPB)
void dist_topk_softmax_kernel(const float* __restrict__ X1,
                              const float* __restrict__ X2,
                              float* __restrict__ out_dense,
                              float* __restrict__ out_score) {
    __shared__ float sq1[ROWS_PER_BLK];
    __shared__ float sq2[CHUNK];
    __shared__ float dist[ROWS_PER_BLK * DPITCH];

    const int tid  = threadIdx.x;
    const int lane = tid & 31;
    const int w    = tid >> 5;
    const int row0 = blockIdx.x * ROWS_PER_BLK;

    // ---- 1) Zero this block's 16-row slab of the dense output (b128 stores).
    {
        float4 z = make_float4(0.f, 0.f, 0.f, 0.f);
        float4* po = (float4*)(out_dense + (size_t)row0 * N2);
        for (int i = tid; i < ROWS_PER_BLK * N2 / 4; i += TPB) po[i] = z;
        // Ensure zeros are committed at L2 before any wave's later scatter.
        asm volatile("s_wait_storecnt 0" ::: "memory");
    }

    // ---- 2) Row norms of X1 slab into LDS.
    if (tid < ROWS_PER_BLK) {
        const float4* p = (const float4*)(X1 + (size_t)(row0 + tid) * DIMK);
        float s = 0.f;
#pragma unroll
        for (int q = 0; q < DIMK / 4; ++q) {
            float4 x = p[q];
            s = fmaf(x.x, x.x, fmaf(x.y, x.y, fmaf(x.z, x.z, fmaf(x.w, x.w, s))));
        }
        sq1[tid] = s;
    }

    // ---- 3) Preload A operand (16x4 f32 WMMA layout): lane = (hi, m),
    // VGPR pair holds K = {2*hi, 2*hi+1} of each 4-wide K step.
    const int n  = lane & 15;
    const int hi = lane >> 4;
    v2f aK[16];
    {
        const float* a_row = X1 + (size_t)(row0 + n) * DIMK + 2 * hi;
#pragma unroll
        for (int kk = 0; kk < 16; ++kk) aK[kk] = *(const v2f*)(a_row + kk * 4);
    }

    // Resident top-32 lists for this wave's two rows (one element per lane).
    float rv[2] = {-3.402823466e38f, -3.402823466e38f};
    int   ri[2] = {0, 0};

    // ---- 4) Column chunks.
    for (int cb = 0; cb < N2; cb += CHUNK) {
        __syncthreads();   // previous chunk's selection done; dist/sq2 reusable

        // Column norms for this chunk.
        for (int c = tid; c < CHUNK; c += TPB) {
            const float4* p = (const float4*)(X2 + (size_t)(cb + c) * DIMK);
            float s = 0.f;
#pragma unroll
            for (int q = 0; q < DIMK / 4; ++q) {
                float4 x = p[q];
                s = fmaf(x.x, x.x, fmaf(x.y, x.y, fmaf(x.z, x.z, fmaf(x.w, x.w, s))));
            }
            sq2[c] = s;
        }
        __syncthreads();

        // WMMA tiles: G = X1_slab * X2_chunk^T, 16 x (16x16x4) per tile.
        for (int t = 0; t < TILES_PER_WAVE; ++t) {
            const int colbase = cb + (w * TILES_PER_WAVE + t) * 16;
            const float* b_row = X2 + (size_t)(colbase + n) * DIMK + 2 * hi;
            __builtin_prefetch(b_row + 16 * DIMK, 0, 1);  // next tile's B rows

            // Batch all 16 B-operand loads so they issue back-to-back and the
            // wave takes ONE L2 latency per tile instead of 16 serialized
            // load->wait->wmma chains.
            v2f bK[16];
#pragma unroll
            for (int kk = 0; kk < 16; ++kk) bK[kk] = *(const v2f*)(b_row + kk * 4);

            // Two accumulators break the serial D->C dependency chain.
            v8f acc0 = {};
            v8f acc1 = {};
#pragma unroll
            for (int kk = 0; kk < 16; kk += 2) {
                acc0 = __builtin_amdgcn_wmma_f32_16x16x4_f32(
                    false, aK[kk],     false, bK[kk],     (short)0, acc0, false, false);
                acc1 = __builtin_amdgcn_wmma_f32_16x16x4_f32(
                    false, aK[kk + 1], false, bK[kk + 1], (short)0, acc1, false, false);
            }

            const float s2 = sq2[(colbase - cb) + n];
#pragma unroll
            for (int v = 0; v < 8; ++v) {
                const int cm = v + 8 * hi;               // C/D layout row
                float g  = acc0[v] + acc1[v];
                float d2 = sq1[cm] + s2 - 2.0f * g;
                float nd = -sqrtf(fmaxf(d2, 0.0f));
                dist[cm * DPITCH + (colbase - cb) + n] = nd;
            }
        }
        __syncthreads();

        // In-register bitonic top-32 maintenance: wave w owns rows 2w, 2w+1.
#pragma unroll
        for (int rr = 0; rr < 2; ++rr) {
            const int rloc = 2 * w + rr;
            for (int b = 0; b < CHUNK / 32; ++b) {
                float bv = dist[rloc * DPITCH + b * 32 + lane];
                int   bi = cb + b * 32 + lane;
                const float rmin = __shfl(rv[rr], 31, 32);
                if (__any(bv > rmin)) {
                    bitonic_sort_asc(bv, bi);
                    topk_merge_desc(rv[rr], ri[rr], bv, bi);
                }
            }
        }
    }

    // ---- 5) Softmax over the sorted top-32, scatter into dense + score rows.
#pragma unroll
    for (int rr = 0; rr < 2; ++rr) {
        const int grow = row0 + 2 * w + rr;
        const float vmax = __shfl(rv[rr], 0, 32);       // list is descending
        const float e = __expf(rv[rr] - vmax);
        float s = e;
#pragma unroll
        for (int off = 16; off >= 1; off >>= 1) s += __shfl_xor(s, off, 32);
        const float sc = e / s;
        out_dense[(size_t)grow * N2 + ri[rr]] = sc;      // scatter (zeros elsewhere)
        out_score[(size_t)grow * TOPK + lane] = sc;      // sorted scores
    }
}

extern "C" void kernel_launch(void* const* d_in, const int* in_sizes, int n_in,
                              void* d_out, int out_size, void* d_ws, size_t ws_size,
                              hipStream_t stream) {
    (void)in_sizes; (void)n_in; (void)out_size; (void)d_ws; (void)ws_size;
    const float* X1 = (const float*)d_in[0];
    const float* X2 = (const float*)d_in[1];
    float* out_dense = (float*)d_out;
    float* out_score = out_dense + (size_t)N1 * N2;

    dim3 grid(N1 / ROWS_PER_BLK);
    dim3 block(TPB);
    hipLaunchKernelGGL(dist_topk_softmax_kernel, grid, block, 0, stream,
                       X1, X2, out_dense, out_score);
}